// Mamba3Layer_79688823210386
// MI455X (gfx1250) — compile-verified
//
#include <hip/hip_runtime.h>
#include <hip/hip_bf16.h>
#include <stdint.h>

// ---------------------------------------------------------------------------
// Mamba3 layer for MI455X (gfx1250):
//   f32 WMMA GEMMs; A-tile staging via TDM (tensor_load_to_lds) or per-lane
//   async-to-LDS copies, double buffered; parallel-per-row bilinear SSM scan;
//   fused silu gate; LayerNorm.
//   B=2, T=2048, D=1024, DS=16, DI=2048, M=B*T=4096
// ---------------------------------------------------------------------------

typedef float v2f  __attribute__((ext_vector_type(2)));
typedef float v8f  __attribute__((ext_vector_type(8)));
typedef unsigned int u32x4 __attribute__((ext_vector_type(4)));
typedef int   i32x4 __attribute__((ext_vector_type(4)));
typedef int   i32x8 __attribute__((ext_vector_type(8)));

#define M_ROWS 4096
#define D_IN   1024
#define DI_    2048
#define DS_    16
#define T_     2048
#define B_     2

#define KCHUNK 32          // K elements staged per double-buffer phase
#define APAD   36          // LDS row stride (floats): bank-conflict-free for the
                           // 16-lane fragment reads; 144B rows, 16B-aligned segs

__device__ __forceinline__ void async_copy_b128(uint32_t lds_off, const float* gptr) {
    // GLOBAL_LOAD_ASYNC_TO_LDS_B128: per-lane 16B global->LDS, tracked by ASYNCcnt
    asm volatile("global_load_async_to_lds_b128 %0, %1, off"
                 :: "v"(lds_off), "v"(gptr) : "memory");
}

__device__ __forceinline__ void wait_async(int keep2) {
    if (keep2) asm volatile("s_wait_asynccnt 0x2" ::: "memory");
    else       asm volatile("s_wait_asynccnt 0x0" ::: "memory");
}

// TDM: load a 64-row x 32-float tile of X (row stride K floats) into LDS with
// a 4-DWORD pad after every 32 DWORDs -> padded APAD=36 row layout.
// D# packing per CDNA5 ISA.
__device__ __forceinline__ void tdm_issue_tile(uint32_t lds_addr, const float* gptr, int K) {
    const uint64_t ga = (uint64_t)(uintptr_t)gptr;
    u32x4 g0 = {
        1u,                                  // count=1, is_restore=0, gather off
        lds_addr,                            // lds_addr [63:32]
        (uint32_t)ga,                        // global_addr[31:0]
        (uint32_t)((ga >> 32) & 0x01FFFFFFu) // global_addr[56:32]
            | 0x80000000u                    // type=2 ("image") in [127:126]
    };
    i32x8 g1 = {
        (int)((2u << 16)        // data_size = 4B
            | (1u << 20)        // pad_enable
            | (4u << 22)        // pad_interval: 32 DWORDs
            | (3u << 25)),      // pad_amount:   4 DWORDs
        (int)((K & 0xFFFF) << 16),                        // tensor_dim0[15:0]
        (int)(((K >> 16) & 0xFFFF) | (4096 << 16)),       // td0 hi | tensor_dim1 lo
        (int)(32 << 16),                                  // td1 hi=0 | tile_dim0=32
        64,                                               // tile_dim1=64, tile_dim2=0
        K,                                                // tensor_dim0_stride[31:0]
        0, 0
    };
    i32x4 z4 = { 0, 0, 0, 0 };
    i32x8 z8 = { 0, 0, 0, 0, 0, 0, 0, 0 };
    // 6-arg variant (clang-23 / therock headers): (g0, g1, g2, g3, extra, cpol)
    __builtin_amdgcn_tensor_load_to_lds(g0, g1, z4, z4, z8, 0);
}

// Out[m][n] = sum_k X[m][k] * W[n][k]   (X: MxK, W: NxK row-major)
// Block = 8 waves sharing a 64-row A block staged in LDS (double buffered via
// TDM when USE_TDM, else per-lane async copies). Each wave: 64x64 macro-tile
// = 4x4 grid of 16x16 WMMA tiles. Requires M%64==0, N%512==0-per-grid, K%32==0.
template <bool USE_TDM>
__global__ __launch_bounds__(256) void gemm_xwT_wmma_big(
    const float* __restrict__ X, const float* __restrict__ W,
    float* __restrict__ Out, int N, int K)
{
    __shared__ float sA[2][64 * APAD];

    const int lane = threadIdx.x;          // 0..31 (wave32)
    const int wave = threadIdx.y;          // 0..7
    const int tid  = wave * 32 + lane;
    const int m0 = blockIdx.x * 64;
    const int n0 = (blockIdx.y * 8 + wave) * 64;

    const int l15 = lane & 15;
    const int hi  = lane >> 4;             // K-pair select / upper M half

    const uint32_t ldsBase  = (uint32_t)(size_t)(&sA[0][0]);
    const uint32_t bufBytes = (uint32_t)(64 * APAD * 4);

    // per-lane async staging plan (used when !USE_TDM):
    // 64 rows x 32 k = 512 B128 segments, 2/thread
    const int r0 = tid >> 3,         kg0 = (tid & 7) * 4;         // rows 0..31
    const int r1 = (tid + 256) >> 3, kg1 = (tid & 7) * 4;         // rows 32..63
    const float* gA0 = X + (size_t)(m0 + r0) * K + kg0;
    const float* gA1 = X + (size_t)(m0 + r1) * K + kg1;
    const uint32_t lOff0 = (uint32_t)((r0 * APAD + kg0) * 4);
    const uint32_t lOff1 = (uint32_t)((r1 * APAD + kg1) * 4);

    // TDM tile origin (used when USE_TDM)
    const float* gTile = X + (size_t)m0 * K;

    // B fragment pointers: B[k][n] = W[n][k]; per-lane float2 at [n0+nt*16+l15][k+2*hi]
    const float* Wp0 = W + (size_t)(n0 +  0 + l15) * K + 2 * hi;
    const float* Wp1 = W + (size_t)(n0 + 16 + l15) * K + 2 * hi;
    const float* Wp2 = W + (size_t)(n0 + 32 + l15) * K + 2 * hi;
    const float* Wp3 = W + (size_t)(n0 + 48 + l15) * K + 2 * hi;

    v8f acc[4][4];
#pragma unroll
    for (int mt = 0; mt < 4; ++mt)
#pragma unroll
        for (int nt = 0; nt < 4; ++nt) acc[mt][nt] = (v8f){};

    // prologue: stage chunk 0
    if (USE_TDM) {
        if (wave == 0) tdm_issue_tile(ldsBase, gTile, K);
    } else {
        async_copy_b128(ldsBase + lOff0, gA0);
        async_copy_b128(ldsBase + lOff1, gA1);
    }

    const int nch = K / KCHUNK;
    for (int ch = 0; ch < nch; ++ch) {
        const int kbase = ch * KCHUNK;
        if (USE_TDM) {
            // one wave drives the DMA; TDM ops from one wave retire in order,
            // the barrier below publishes completion to the other 7 waves
            if (wave == 0) {
                if (ch + 1 < nch) {
                    const uint32_t nxt = ldsBase + (uint32_t)((ch + 1) & 1) * bufBytes;
                    tdm_issue_tile(nxt, gTile + kbase + KCHUNK, K);
                    __builtin_amdgcn_s_wait_tensorcnt(1);  // chunk ch resident
                } else {
                    __builtin_amdgcn_s_wait_tensorcnt(0);
                }
            }
            if (ch + 1 < nch) __builtin_prefetch(Wp0 + kbase + 2 * KCHUNK, 0, 0);
        } else {
            if (ch + 1 < nch) {
                const uint32_t nxt = ldsBase + (uint32_t)((ch + 1) & 1) * bufBytes;
                async_copy_b128(nxt + lOff0, gA0 + kbase + KCHUNK);
                async_copy_b128(nxt + lOff1, gA1 + kbase + KCHUNK);
                __builtin_prefetch(Wp0 + kbase + 2 * KCHUNK, 0, 0);
                wait_async(1);      // async loads retire in order -> chunk ch resident
            } else {
                wait_async(0);
            }
        }
        __syncthreads();

        const float* Ab = &sA[ch & 1][0];
#pragma unroll
        for (int kk = 0; kk < KCHUNK; kk += 4) {
            v2f a0 = *(const v2f*)(Ab + ( 0 + l15) * APAD + kk + 2 * hi);
            v2f a1 = *(const v2f*)(Ab + (16 + l15) * APAD + kk + 2 * hi);
            v2f a2 = *(const v2f*)(Ab + (32 + l15) * APAD + kk + 2 * hi);
            v2f a3 = *(const v2f*)(Ab + (48 + l15) * APAD + kk + 2 * hi);
            v2f b0 = *(const v2f*)(Wp0 + kbase + kk);
            v2f b1 = *(const v2f*)(Wp1 + kbase + kk);
            v2f b2 = *(const v2f*)(Wp2 + kbase + kk);
            v2f b3 = *(const v2f*)(Wp3 + kbase + kk);
#define WMMA4(mt, areg)                                                                          \
            acc[mt][0] = __builtin_amdgcn_wmma_f32_16x16x4_f32(false, areg, false, b0, (short)0, acc[mt][0], false, false); \
            acc[mt][1] = __builtin_amdgcn_wmma_f32_16x16x4_f32(false, areg, false, b1, (short)0, acc[mt][1], false, false); \
            acc[mt][2] = __builtin_amdgcn_wmma_f32_16x16x4_f32(false, areg, false, b2, (short)0, acc[mt][2], false, false); \
            acc[mt][3] = __builtin_amdgcn_wmma_f32_16x16x4_f32(false, areg, false, b3, (short)0, acc[mt][3], false, false);
            WMMA4(0, a0) WMMA4(1, a1) WMMA4(2, a2) WMMA4(3, a3)
#undef WMMA4
        }
        __syncthreads();   // protect buffer ch&1 before next issue overwrites
    }

    // store: D layout -> VGPR i: M = mtile + i + 8*hi, N = ntile + l15
    const int rbase = hi * 8;
#pragma unroll
    for (int mt = 0; mt < 4; ++mt)
#pragma unroll
        for (int i = 0; i < 8; ++i) {
            const size_t row = (size_t)(m0 + mt * 16 + rbase + i) * N;
            Out[row + n0 +  0 + l15] = acc[mt][0][i];
            Out[row + n0 + 16 + l15] = acc[mt][1][i];
            Out[row + n0 + 32 + l15] = acc[mt][2][i];
            Out[row + n0 + 48 + l15] = acc[mt][3][i];
        }
}

// N==16 GEMM (WB / WC projections): one n-tile, zero predication.
// Each wave: 32 rows x 16 cols. Block = 8 waves = 256 rows.
__global__ __launch_bounds__(256) void gemm_xwT_wmma_n16(
    const float* __restrict__ X, const float* __restrict__ W,
    float* __restrict__ Out, int K)
{
    const int lane = threadIdx.x;
    const int wave = threadIdx.y;
    const int m0 = (blockIdx.x * 8 + wave) * 32;
    const int l15 = lane & 15;
    const int hi  = lane >> 4;

    const float* Xp0 = X + (size_t)(m0 + l15) * K + 2 * hi;
    const float* Xp1 = X + (size_t)(m0 + 16 + l15) * K + 2 * hi;
    const float* Wp  = W + (size_t)l15 * K + 2 * hi;

    v8f c0 = (v8f){}, c1 = (v8f){};
    for (int k = 0; k < K; k += 4) {
        v2f a0 = *(const v2f*)(Xp0 + k);
        v2f a1 = *(const v2f*)(Xp1 + k);
        v2f b  = *(const v2f*)(Wp + k);
        c0 = __builtin_amdgcn_wmma_f32_16x16x4_f32(false, a0, false, b, (short)0, c0, false, false);
        c1 = __builtin_amdgcn_wmma_f32_16x16x4_f32(false, a1, false, b, (short)0, c1, false, false);
    }
    const int rbase = hi * 8;
#pragma unroll
    for (int i = 0; i < 8; ++i) {
        Out[(size_t)(m0 + rbase + i) * DS_ + l15]      = c0[i];
        Out[(size_t)(m0 + 16 + rbase + i) * DS_ + l15] = c1[i];
    }
}

// Sequential scan over T; fully parallel across 4096 (b,di) rows.
// xz: (M,4096): cols [0,2048)=xi, [2048,4096)=z-pre. out: (M, DI).
__global__ __launch_bounds__(256) void mamba_scan(
    const float* __restrict__ xz, const float* __restrict__ dtpre,
    const float* __restrict__ Bm, const float* __restrict__ Cm,
    const float* __restrict__ A_log,
    const float* __restrict__ bdt, const float* __restrict__ dt_bias,
    float* __restrict__ out, float* __restrict__ state_out)
{
    const int gid = blockIdx.x * blockDim.x + threadIdx.x;  // 0..4095
    const int b  = gid / DI_;          // uniform per block (256 | 2048)
    const int di = gid % DI_;

    __shared__ float sB[DS_];
    __shared__ float sC[DS_];

    float Arow[DS_];
#pragma unroll
    for (int s = 0; s < DS_; ++s) Arow[s] = -expf(A_log[(size_t)di * DS_ + s]);
    const float dtb = bdt[di] + dt_bias[di];

    float st[DS_];
#pragma unroll
    for (int s = 0; s < DS_; ++s) st[s] = 0.f;

    for (int t = 0; t < T_; ++t) {
        const int idx = b * T_ + t;
        if (threadIdx.x < DS_)            sB[threadIdx.x]       = Bm[(size_t)idx * DS_ + threadIdx.x];
        else if (threadIdx.x < 2 * DS_)   sC[threadIdx.x - DS_] = Cm[(size_t)idx * DS_ + threadIdx.x - DS_];
        __syncthreads();

        const float dp = dtpre[(size_t)idx * DI_ + di] + dtb;
        const float dt = (dp > 20.f) ? dp : log1pf(expf(dp));   // softplus
        const float xi = xz[(size_t)idx * (2 * DI_) + di];
        const float zp = xz[(size_t)idx * (2 * DI_) + DI_ + di];
        const float z  = zp / (1.f + expf(-zp));                // silu

        float accv = 0.f;
#pragma unroll
        for (int s = 0; s < DS_; ++s) {
            const float dtA   = dt * Arow[s];
            const float denom = 1.f - 0.5f * dtA;
            const float Ab2   = (1.f + 0.5f * dtA) / denom;     // bilinear
            const float Bb    = dt / denom;
            st[s] = Ab2 * st[s] + Bb * sB[s] * xi;
            accv += st[s] * sC[s];
        }
        out[(size_t)idx * DI_ + di] = accv * z;
        __syncthreads();
    }
#pragma unroll
    for (int s = 0; s < DS_; ++s) state_out[(size_t)gid * DS_ + s] = st[s];
}

// In-place LayerNorm over DI per row.
__global__ __launch_bounds__(256) void layernorm_inplace(
    float* __restrict__ buf, const float* __restrict__ gamma,
    const float* __restrict__ beta)
{
    float* p = buf + (size_t)blockIdx.x * DI_;
    __shared__ float red[256];

    float s1 = 0.f, s2 = 0.f;
    for (int i = threadIdx.x; i < DI_; i += 256) {
        const float v = p[i];
        s1 += v; s2 += v * v;
    }
    red[threadIdx.x] = s1; __syncthreads();
    for (int off = 128; off > 0; off >>= 1) {
        if (threadIdx.x < off) red[threadIdx.x] += red[threadIdx.x + off];
        __syncthreads();
    }
    const float mean = red[0] / (float)DI_;
    __syncthreads();
    red[threadIdx.x] = s2; __syncthreads();
    for (int off = 128; off > 0; off >>= 1) {
        if (threadIdx.x < off) red[threadIdx.x] += red[threadIdx.x + off];
        __syncthreads();
    }
    const float var = red[0] / (float)DI_ - mean * mean;
    const float inv = rsqrtf(var + 1e-5f);
    for (int i = threadIdx.x; i < DI_; i += 256)
        p[i] = (p[i] - mean) * inv * gamma[i] + beta[i];
}

extern "C" void kernel_launch(void* const* d_in, const int* in_sizes, int n_in,
                              void* d_out, int out_size, void* d_ws, size_t ws_size,
                              hipStream_t stream) {
    (void)in_sizes; (void)n_in; (void)out_size; (void)ws_size;
    const float* x       = (const float*)d_in[0];   // (B,T,D)
    const float* Win     = (const float*)d_in[1];   // (2*DI, D)
    const float* A_log   = (const float*)d_in[2];   // (DI, DS)
    const float* WB      = (const float*)d_in[3];   // (DS, D)
    const float* WC      = (const float*)d_in[4];   // (DS, D)
    const float* Wdt     = (const float*)d_in[5];   // (DI, D)
    const float* bdt     = (const float*)d_in[6];   // (DI)
    const float* dt_bias = (const float*)d_in[7];   // (DI)
    const float* gamma   = (const float*)d_in[8];   // (DI)
    const float* beta    = (const float*)d_in[9];   // (DI)
    const float* Wout    = (const float*)d_in[10];  // (D, DI)

    float* ws   = (float*)d_ws;
    float* xz   = ws;                                   // M x 4096
    float* dtp  = xz  + (size_t)M_ROWS * (2 * DI_);     // M x 2048
    float* obuf = dtp + (size_t)M_ROWS * DI_;           // M x 2048
    float* BmB  = obuf + (size_t)M_ROWS * DI_;          // M x 16
    float* CmB  = BmB  + (size_t)M_ROWS * DS_;          // M x 16

    float* y_out     = (float*)d_out;                       // (B,T,D)
    float* state_out = y_out + (size_t)M_ROWS * D_IN;       // (B,DI,DS)

    dim3 blk(32, 8);
    // xz = x @ Win^T   (N=4096, K=1024)  -- TDM-staged A
    gemm_xwT_wmma_big<true><<<dim3(M_ROWS / 64, (2 * DI_) / 512), blk, 0, stream>>>(
        x, Win, xz, 2 * DI_, D_IN);
    // dtpre = x @ Wdt^T (N=2048, K=1024) -- TDM-staged A
    gemm_xwT_wmma_big<true><<<dim3(M_ROWS / 64, DI_ / 512), blk, 0, stream>>>(
        x, Wdt, dtp, DI_, D_IN);
    // Bm = x @ WB^T, C = x @ WC^T (N=16)
    gemm_xwT_wmma_n16<<<dim3(M_ROWS / 256), blk, 0, stream>>>(x, WB, BmB, D_IN);
    gemm_xwT_wmma_n16<<<dim3(M_ROWS / 256), blk, 0, stream>>>(x, WC, CmB, D_IN);
    // scan: 4096 independent rows (16 blocks x 256 threads)
    mamba_scan<<<dim3((B_ * DI_) / 256), dim3(256), 0, stream>>>(
        xz, dtp, BmB, CmB, A_log, bdt, dt_bias, obuf, state_out);
    // layernorm in-place on obuf
    layernorm_inplace<<<dim3(M_ROWS), dim3(256), 0, stream>>>(obuf, gamma, beta);
    // y = normed @ Wout^T (N=1024, K=2048) -- async-copy-staged A
    gemm_xwT_wmma_big<false><<<dim3(M_ROWS / 64, D_IN / 512), blk, 0, stream>>>(
        obuf, Wout, y_out, D_IN, DI_);
}